// DeformConv3d_10934986736277
// MI455X (gfx1250) — compile-verified
//
#include <hip/hip_runtime.h>

typedef __attribute__((ext_vector_type(2))) float v2f;
typedef __attribute__((ext_vector_type(8))) float v8f;

#define CIN    16
#define NPTS   27            // 3^3 sampling taps
#define KDIM   432           // CIN * NPTS
#define NOFF   81            // 3 * NPTS offset channels
#define SDIM   32
#define VOX    32768         // 32^3
#define MT     16            // voxels per block (WMMA M tile)
#define ASTR   433           // LDS row stride (odd -> bank-conflict free)
#define OSTR   84            // offset LDS row stride
#define KSTEPS 108           // KDIM / 4 (K per V_WMMA_F32_16X16X4_F32)

// padded-x read: xp[c][a][b][d], a/b/d in [0,33], zero ring of width 1
__device__ __forceinline__ float load_xpad(const float* __restrict__ x,
                                           int c, int a, int b, int d) {
    bool in = ((unsigned)(a - 1) < (unsigned)SDIM) &
              ((unsigned)(b - 1) < (unsigned)SDIM) &
              ((unsigned)(d - 1) < (unsigned)SDIM);
    int idx = ((c * SDIM + (a - 1)) * SDIM + (b - 1)) * SDIM + (d - 1);
    return in ? x[idx] : 0.0f;
}

// One 16x16 fp32 WMMA output tile: D = A(16xKDIM, LDS) * B(KDIMx16) + 0
// B row for this lane pre-offset by caller: Brow = B + n*KDIM, bscale masks padded n.
__device__ __forceinline__ v8f wmma_tile_f32(const float* __restrict__ Alds,
                                             const float* __restrict__ Brow,
                                             int lane, float bscale) {
    const int mrow = lane & 15;
    const int koff = (lane >> 4) << 1;          // lanes 0-15: K {0,1}; 16-31: K {2,3}
    const float* ap = Alds + mrow * ASTR + koff;
    const float* bp = Brow + koff;
    v8f acc = {};
#pragma unroll 4
    for (int kk = 0; kk < KSTEPS; ++kk) {
        v2f a, b;
        a.x = ap[0];           a.y = ap[1];
        b.x = bp[0] * bscale;  b.y = bp[1] * bscale;
        ap += 4;               bp += 4;
        acc = __builtin_amdgcn_wmma_f32_16x16x4_f32(
            /*neg_a=*/false, a, /*neg_b=*/false, b,
            /*c_mod=*/(short)0, acc, /*reuse_a=*/false, /*reuse_b=*/false);
    }
    return acc;
}

__global__ __launch_bounds__(128)
void deform_conv3d_fused(const float* __restrict__ x,      // (16, 32,32,32)
                         const float* __restrict__ offw,   // (81, 432) row-major
                         const float* __restrict__ convw,  // (64, 432) row-major
                         float* __restrict__ out)          // (64, 32768)
{
    __shared__ float A[MT * ASTR];     // 16 x 432 im2col / x_off tile (reused)
    __shared__ float OFF[MT * OSTR];   // 16 x 81 predicted offsets

    const int tid  = threadIdx.x;
    const int lane = tid & 31;
    const int wave = tid >> 5;         // 0..3
    const int m0   = blockIdx.x * MT;

    // ---------- Phase 1: im2col of x for the offset conv ----------
    for (int p = tid; p < MT * KDIM; p += 128) {
        int m = p / KDIM, k = p - m * KDIM;
        int c = k / NPTS, t = k - c * NPTS;
        int g = m0 + m;
        int i0 = g >> 10, i1 = (g >> 5) & 31, i2 = g & 31;
        // padded coord = (i+1) + (dt-1) = i + dt, dt in {0,1,2}
        A[m * ASTR + k] = load_xpad(x, c, i0 + t / 9, i1 + (t / 3) % 3, i2 + t % 3);
    }
    __syncthreads();

    // ---------- Phase 2: offset GEMM, 16 x 96 (81 valid) ----------
    for (int nt = wave; nt < 6; nt += 4) {
        int   n  = nt * 16 + (lane & 15);
        int   nc = (n < NOFF) ? n : (NOFF - 1);
        float bm = (n < NOFF) ? 1.0f : 0.0f;
        v8f acc = wmma_tile_f32(A, offw + nc * KDIM, lane, bm);
#pragma unroll
        for (int r = 0; r < 8; ++r) {
            int mm = r + ((lane >> 4) << 3);
            if (n < NOFF) OFF[mm * OSTR + n] = acc[r];
        }
    }
    __syncthreads();

    // ---------- Phase 3: trilinear gather -> x_off tile (overwrites A) ----------
    for (int p = tid; p < MT * NPTS; p += 128) {
        int m = p / NPTS, t = p - m * NPTS;
        int g = m0 + m;
        int i0 = g >> 10, i1 = (g >> 5) & 31, i2 = g & 31;

        float px = (float)(i0 + t / 9)       + OFF[m * OSTR + t];
        float py = (float)(i1 + (t / 3) % 3) + OFF[m * OSTR + NPTS + t];
        float pz = (float)(i2 + t % 3)       + OFF[m * OSTR + 2 * NPTS + t];

        float fx = floorf(px), fy = floorf(py), fz = floorf(pz);
        // corners from RAW p (reference order), clamped to [0, 33]
        int qx[2], qy[2], qz[2];
        qx[0] = (int)fminf(fmaxf(fx, 0.f), 33.f);
        qx[1] = (int)fminf(fmaxf(fx + 1.f, 0.f), 33.f);
        qy[0] = (int)fminf(fmaxf(fy, 0.f), 33.f);
        qy[1] = (int)fminf(fmaxf(fy + 1.f, 0.f), 33.f);
        qz[0] = (int)fminf(fmaxf(fz, 0.f), 33.f);
        qz[1] = (int)fminf(fmaxf(fz + 1.f, 0.f), 33.f);

        // mask fix: snap out-of-range p to floor, then clamp
        if (px < 1.f || px > 32.f) px = fx;
        if (py < 1.f || py > 32.f) py = fy;
        if (pz < 1.f || pz > 32.f) pz = fz;
        px = fminf(fmaxf(px, 0.f), 33.f);
        py = fminf(fmaxf(py, 0.f), 33.f);
        pz = fminf(fmaxf(pz, 0.f), 33.f);

        float gx[2], gy[2], gz[2];
        gx[0] = 1.f + ((float)qx[0] - px);  gx[1] = 1.f - ((float)qx[1] - px);
        gy[0] = 1.f + ((float)qy[0] - py);  gy[1] = 1.f - ((float)qy[1] - py);
        gz[0] = 1.f + ((float)qz[0] - pz);  gz[1] = 1.f - ((float)qz[1] - pz);

        float wgt[8];
        int   base[8];
#pragma unroll
        for (int j = 0; j < 8; ++j) {
            int a = qx[j >> 2], b = qy[(j >> 1) & 1], d = qz[j & 1];
            bool in = ((unsigned)(a - 1) < 32u) & ((unsigned)(b - 1) < 32u) &
                      ((unsigned)(d - 1) < 32u);
            // pad ring of xp is zero -> zero the weight instead of the value
            wgt[j]  = in ? gx[j >> 2] * gy[(j >> 1) & 1] * gz[j & 1] : 0.f;
            base[j] = in ? (((a - 1) * SDIM + (b - 1)) * SDIM + (d - 1)) : 0;
        }

        const float* xc = x;
#pragma unroll 1
        for (int c = 0; c < CIN; ++c) {
            float v = 0.f;
#pragma unroll
            for (int j = 0; j < 8; ++j) v += wgt[j] * xc[base[j]];
            A[m * ASTR + c * NPTS + t] = v;
            xc += VOX;
        }
    }
    __syncthreads();

    // ---------- Phase 4: output GEMM, 16 x 64 ----------
    {
        int n = wave * 16 + (lane & 15);
        v8f acc = wmma_tile_f32(A, convw + n * KDIM, lane, 1.0f);
        float* op = out + n * VOX + m0;
#pragma unroll
        for (int r = 0; r < 8; ++r) {
            int mm = r + ((lane >> 4) << 3);
            op[mm] = acc[r];
        }
    }
}

extern "C" void kernel_launch(void* const* d_in, const int* in_sizes, int n_in,
                              void* d_out, int out_size, void* d_ws, size_t ws_size,
                              hipStream_t stream) {
    (void)in_sizes; (void)n_in; (void)out_size; (void)d_ws; (void)ws_size;
    const float* x     = (const float*)d_in[0];
    const float* offw  = (const float*)d_in[1];   // (81,16,3,3,3) -> (81,432)
    const float* convw = (const float*)d_in[2];   // (64,432,1,1,1) -> (64,432)
    float* out = (float*)d_out;                   // (1,64,32,32,32)

    deform_conv3d_fused<<<VOX / MT, 128, 0, stream>>>(x, offw, convw, out);
}